// TopK_61211873903224
// MI455X (gfx1250) — compile-verified
//
#include <hip/hip_runtime.h>
#include <stdint.h>

// ---------------------------------------------------------------------------
// Per-row top-K (K=64) masking on an 8192x8192 fp32 matrix.
// One 256-thread block per row. Row cached in LDS (32 KB of 320 KB/WGP) via
// CDNA5 async global->LDS B128 DMA. Threshold via 4-pass radix select on
// sign-flipped float keys; bin selection done by a wave32 parallel suffix
// scan (shfl-based, no serial thread-0 loop). Emit pass uses non-temporal
// 128-bit stores. HBM traffic = 1x read + 1x write (~22 us at 23.3 TB/s).
// ---------------------------------------------------------------------------

#define N_COLS 8192
#define K_TOP 64
#define BLOCK 256
#define PER_THREAD (N_COLS / BLOCK) // 32 floats / thread
#define VEC_ITERS (PER_THREAD / 4)  // 8 float4 / thread

// Native clang vector types (HIP_vector_type is rejected by the NT builtin).
typedef float v4f __attribute__((ext_vector_type(4)));
typedef int v4i __attribute__((vector_size(16)));
typedef __attribute__((address_space(1))) v4i as1_v4i; // global ("__device__")
typedef __attribute__((address_space(3))) v4i as3_v4i; // LDS ("__shared__")

// Monotonic key: larger float  <=>  larger unsigned key.
__device__ __forceinline__ uint32_t f2key(float f) {
    uint32_t u = __float_as_uint(f);
    return (u & 0x80000000u) ? ~u : (u | 0x80000000u);
}

__global__ __launch_bounds__(BLOCK) void topk_mask_kernel(
    const float* __restrict__ x, float* __restrict__ out) {
    __shared__ float row[N_COLS];          // 32 KB
    __shared__ uint32_t hist[256];         // 1 KB
    __shared__ uint32_t sel_prefix;        // accumulated key bytes (high->low)
    __shared__ uint32_t sel_rank;          // remaining rank within current prefix
    __shared__ uint32_t tie_counter;

    const int t = threadIdx.x;
    const float* __restrict__ xr = x + (size_t)blockIdx.x * N_COLS;
    float* __restrict__ outr = out + (size_t)blockIdx.x * N_COLS;

    // ---- Phase 0: stage row into LDS (async DMA, ASYNCcnt-tracked) --------
#if defined(__gfx1250__) && __has_builtin(__builtin_amdgcn_global_load_async_to_lds_b128)
    {
        as1_v4i* gsrc = (as1_v4i*)xr;
        as3_v4i* ldst = (as3_v4i*)row;
        #pragma unroll
        for (int j = 0; j < VEC_ITERS; ++j) {
            int idx = t + j * BLOCK; // 16-byte elements
            __builtin_amdgcn_global_load_async_to_lds_b128(
                gsrc + idx, ldst + idx, /*offset=*/0, /*cpol=*/0);
        }
#if __has_builtin(__builtin_amdgcn_s_wait_asynccnt)
        __builtin_amdgcn_s_wait_asynccnt(0);
#else
        asm volatile("s_wait_asynccnt 0" ::: "memory");
#endif
    }
#else
    {
        const v4f* __restrict__ src4 = (const v4f*)xr;
        v4f* dst4 = (v4f*)row;
        #pragma unroll
        for (int j = 0; j < VEC_ITERS; ++j) {
            dst4[t + j * BLOCK] = src4[t + j * BLOCK];
        }
    }
#endif
    if (t == 0) {
        sel_prefix = 0u;
        sel_rank = K_TOP;
        tie_counter = 0u;
    }
    __syncthreads();

    // ---- Phase 1: 4-pass radix select (8 bits / pass, MSB first) ----------
    uint32_t cur_prefix = 0u;
    #pragma unroll
    for (int pass = 0; pass < 4; ++pass) {
        const int shift = 24 - 8 * pass;
        const uint32_t pmask =
            (pass == 0) ? 0u : (0xFFFFFFFFu << (shift + 8));

        hist[t] = 0u; // BLOCK == 256 bins
        __syncthreads();

        for (int i = t; i < N_COLS; i += BLOCK) {
            uint32_t key = f2key(row[i]);
            if ((key & pmask) == cur_prefix) {
                atomicAdd(&hist[(key >> shift) & 0xFFu], 1u);
            }
        }
        __syncthreads();

        // Wave32-parallel bin selection: lane L owns bins [8L, 8L+7].
        // Find the largest bin b with suffixSum(b) >= rank (rank-th largest).
        if (t < 32) {
            const uint32_t rank = sel_rank;
            uint32_t h[8];
            #pragma unroll
            for (int j = 0; j < 8; ++j) h[j] = hist[t * 8 + j];

            // Local inclusive suffix sums within the lane's 8 bins.
            uint32_t ls[8];
            ls[7] = h[7];
            #pragma unroll
            for (int j = 6; j >= 0; --j) ls[j] = h[j] + ls[j + 1];
            const uint32_t lane_total = ls[0];

            // Inclusive suffix scan across lanes (sum over lanes >= this one).
            uint32_t inc = lane_total;
            #pragma unroll
            for (int off = 1; off < 32; off <<= 1) {
                uint32_t v = __shfl_down(inc, off, 32);
                if (t + off < 32) inc += v;
            }
            const uint32_t excl_above = inc - lane_total; // sum over lanes > L

            // S(8L+j) = ls[j] + excl_above ; S(8L+8) == excl_above.
            #pragma unroll
            for (int j = 0; j < 8; ++j) {
                const uint32_t S = ls[j] + excl_above;
                const uint32_t Snext = (j < 7) ? (ls[j + 1] + excl_above)
                                               : excl_above;
                if (S >= rank && Snext < rank) { // unique (lane, j) matches
                    sel_rank = rank - Snext;     // rank within selected bin
                    sel_prefix |= ((uint32_t)(t * 8 + j) << shift);
                }
            }
        }
        __syncthreads();
        cur_prefix = sel_prefix;
    }

    const uint32_t thresh_key = sel_prefix; // key of the K-th largest element
    const uint32_t n_ties_keep = sel_rank;  // #elements equal to thresh to keep

    // ---- Phase 2: emit masked row (coalesced non-temporal 128b stores) ----
    v4f* __restrict__ out4 = (v4f*)outr;
    #pragma unroll
    for (int j = 0; j < VEC_ITERS; ++j) {
        const int base = (t + j * BLOCK) * 4;
        v4f o;
        #pragma unroll
        for (int k = 0; k < 4; ++k) {
            float v = row[base + k];
            uint32_t key = f2key(v);
            float r = 0.0f;
            if (key > thresh_key) {
                r = v;
            } else if (key == thresh_key) {
                uint32_t idx = atomicAdd(&tie_counter, 1u);
                if (idx < n_ties_keep) r = v;
            }
            o[k] = r;
        }
        __builtin_nontemporal_store(o, &out4[t + j * BLOCK]);
    }
}

extern "C" void kernel_launch(void* const* d_in, const int* in_sizes, int n_in,
                              void* d_out, int out_size, void* d_ws, size_t ws_size,
                              hipStream_t stream) {
    (void)in_sizes; (void)n_in; (void)d_ws; (void)ws_size;
    const float* x = (const float*)d_in[0];
    float* out = (float*)d_out;
    const int n_rows = out_size / N_COLS; // 8192
    topk_mask_kernel<<<n_rows, BLOCK, 0, stream>>>(x, out);
}